// ROIPatch_60464549593191
// MI455X (gfx1250) — compile-verified
//
#include <hip/hip_runtime.h>
#include <stdint.h>

// ROI crop: out[n][b][m][c][r][cc] = mask(r<h, cc<w) * fm[n][b][c][y1+r][x1+cc]
// Shapes: fm (2,8,32,128,128) f32, boxes (2,8,32,4) i32, out (2,8,32,32,64,64) f32.
// Pure masked gather (0 FLOPs). HBM floor ~320MB -> ~14us @ 23.3 TB/s.
// CDNA5 path: stage tile in LDS, drain with global_store_async_from_lds_b128
// (ASYNCcnt), so the store data path bypasses VGPRs and is 16B-vectorized.

#define NS 2
#define BN 8
#define CN 32
#define HN 128
#define WN 128
#define MN 32
#define RH 64
#define RW 64

__global__ __launch_bounds__(256) void roi_crop_async_kernel(
    const float* __restrict__ fm, const int* __restrict__ boxes,
    float* __restrict__ out)
{
    __shared__ float tile[RH * RW];   // 16 KB per block (320 KB/WGP available)

    const int bid = blockIdx.x;                 // ((n*BN+b)*MN+m)*CN + c
    const int c = bid & (CN - 1);
    const int m = (bid >> 5) & (MN - 1);
    const int b = (bid >> 10) & (BN - 1);
    const int n = bid >> 13;

    // Box (block-uniform -> scalar loads)
    const int* bx = boxes + (((n * BN + b) * MN + m) << 2);
    const int x1 = bx[0];
    const int y1 = bx[1];
    const int w  = bx[2] - x1;
    const int h  = bx[3] - y1;
    // Defensive clamp: guarantees all gathers stay inside the 128x128 plane.
    const int xb = min(max(x1, 0), WN - RW);

    const float* plane = fm + (size_t)((n * BN + b) * CN + c) * (size_t)(HN * WN);
    float* obase = out + (size_t)bid * (size_t)(RH * RW);

    const int t = threadIdx.x;

    // Stage 1: gather + mask into LDS. Each thread owns 4 float4 chunks
    // (chunk ids t, t+256, t+512, t+768) and later drains exactly those,
    // so only a per-wave s_wait_dscnt is needed (no barrier).
#pragma unroll
    for (int k = 0; k < 4; ++k) {
        const int q  = t + (k << 8);        // chunk id 0..1023
        const int r  = q >> 4;              // row 0..63
        const int cc = (q & 15) << 2;       // col 0,4,...,60
        const int y  = min(max(y1 + r, 0), HN - 1);
        const float* src = plane + y * WN + xb + cc;  // 4B-aligned (x1 arbitrary)
        const float a0 = src[0];
        const float a1 = src[1];
        const float a2 = src[2];
        const float a3 = src[3];
        const bool rv = (r < h);
        float4 v;
        v.x = (rv && (cc + 0) < w) ? a0 : 0.0f;
        v.y = (rv && (cc + 1) < w) ? a1 : 0.0f;
        v.z = (rv && (cc + 2) < w) ? a2 : 0.0f;
        v.w = (rv && (cc + 3) < w) ? a3 : 0.0f;
        *reinterpret_cast<float4*>(&tile[q << 2]) = v;   // ds_store_b128
    }

    // Ensure this wave's LDS writes have landed before the async engine reads.
    asm volatile("s_wait_dscnt 0x0" ::: "memory");

    // Stage 2: async drain LDS -> global (16B chunks, fully coalesced;
    // output tile base is 16KB-aligned).
#pragma unroll
    for (int k = 0; k < 4; ++k) {
        const int q = t + (k << 8);
        const uint64_t gdst = (uint64_t)(uintptr_t)(obase + (q << 2));
        // Low 32 bits of the flat shared pointer == LDS byte offset.
        const uint32_t lsrc = (uint32_t)(uintptr_t)(&tile[q << 2]);
        asm volatile("global_store_async_from_lds_b128 %0, %1, off"
                     :: "v"(gdst), "v"(lsrc)
                     : "memory");
    }

    // Drain ASYNCcnt (s_endpgm would also wait-idle, but be explicit).
    asm volatile("s_wait_asynccnt 0x0" ::: "memory");
}

extern "C" void kernel_launch(void* const* d_in, const int* in_sizes, int n_in,
                              void* d_out, int out_size, void* d_ws, size_t ws_size,
                              hipStream_t stream) {
    const float* fm    = (const float*)d_in[0];  // (2,8,32,128,128) f32
    const int*   boxes = (const int*)d_in[1];    // (2,8,32,4) i32
    float*       out   = (float*)d_out;          // (2,8,32,32,64,64) f32

    const int nblocks = NS * BN * MN * CN;       // 16384 tiles
    roi_crop_async_kernel<<<nblocks, 256, 0, stream>>>(fm, boxes, out);
}